// GWTNet_layer_22411139351116
// MI455X (gfx1250) — compile-verified
//
#include <hip/hip_runtime.h>
#include <cmath>

// ---------------------------------------------------------------------------
// GWTNet (Graph-WaveNet) forward on gfx1250.
// Internal layout for all activations: [b][n][l(64,padded)][c(32)], c fastest.
// All channel contractions + graph diffusion run through V_WMMA_F32_16X16X4_F32.
// Diffusion stages the adj^T slab through LDS (conflict-free padded pitch).
// ---------------------------------------------------------------------------

typedef __attribute__((ext_vector_type(2))) float v2f;
typedef __attribute__((ext_vector_type(8))) float v8f;

#define EPSBN 1e-5f

__device__ __forceinline__ v8f wmma_f32(v2f a, v2f b, v8f c) {
  // 8 args: (neg_a, A, neg_b, B, c_mod, C, reuse_a, reuse_b)
  return __builtin_amdgcn_wmma_f32_16x16x4_f32(false, a, false, b, (short)0, c,
                                               false, false);
}

// ---------------------------------------------------------------------------
// Kernel 1: start 1x1 conv (IN_DIM=2 -> 32ch), layout change to [b][n][l][c].
// ---------------------------------------------------------------------------
__global__ void k_start(const float* __restrict__ x, const float* __restrict__ Wst,
                        const float* __restrict__ bst, float* __restrict__ H) {
  int t = blockIdx.x * blockDim.x + threadIdx.x;   // exact: 8*512*64*32 threads
  int o   = t & 31;
  int pos = t >> 5;                // (b*512+n)*64 + l
  int l   = pos & 63;
  int n   = (pos >> 6) & 511;
  int b   = pos >> 15;
  const float* xb = x + (size_t)b * 2 * 512 * 64;
  float x0 = xb[(size_t)n * 64 + l];
  float x1 = xb[(size_t)512 * 64 + (size_t)n * 64 + l];
  H[t] = fmaf(x0, Wst[o * 2], fmaf(x1, Wst[o * 2 + 1], bst[o]));
}

// ---------------------------------------------------------------------------
// Kernel 2: gated temporal conv. K=64 (two taps contiguous), two accumulators
// per wave (filter & gate share the A tile), tanh*sigmoid epilogue.
// ---------------------------------------------------------------------------
__global__ void __launch_bounds__(256) k_gated(
    const float* __restrict__ H, float* __restrict__ G,
    const float* __restrict__ Wf, const float* __restrict__ bf,
    const float* __restrict__ Wg, const float* __restrict__ bg, int Lout) {
  int wid  = (blockIdx.x * blockDim.x + threadIdx.x) >> 5;
  int lane = threadIdx.x & 31;
  int half = lane >> 4;
  int l15  = lane & 15;
  int otile = wid & 1;
  int rtile = wid >> 1;
  int m0 = rtile << 4;
  int o  = (otile << 4) + l15;

  const float* Arow = H + (size_t)(m0 + l15) * 32;   // A lane holds M = l15
  float biasf = bf[o], biasg = bg[o];
  v8f cf, cg;
#pragma unroll
  for (int r = 0; r < 8; ++r) { cf[r] = biasf; cg[r] = biasg; }

#pragma unroll
  for (int k0 = 0; k0 < 64; k0 += 4) {
    int k = k0 + half * 2;                  // this lane's K pair: k, k+1
    v2f a = *(const v2f*)(Arow + k);        // taps contiguous in layout
    int c0 = k & 31, tap = k >> 5;
    const float* wf = Wf + (size_t)o * 64 + c0 * 2 + tap;
    const float* wg = Wg + (size_t)o * 64 + c0 * 2 + tap;
    v2f bfv, bgv;
    bfv.x = wf[0]; bfv.y = wf[2];
    bgv.x = wg[0]; bgv.y = wg[2];
    cf = wmma_f32(a, bfv, cf);
    cg = wmma_f32(a, bgv, cg);
  }
#pragma unroll
  for (int r = 0; r < 8; ++r) {
    int row = m0 + r + half * 8;
    int l = row & 63;
    if (l < Lout) {
      float fv = tanhf(cf[r]);
      float gv = 1.0f / (1.0f + __expf(-cg[r]));
      G[(size_t)row * 32 + o] = fv * gv;
    }
  }
}

// ---------------------------------------------------------------------------
// Kernel 3: skip 1x1 conv, accumulating into the final 61-length skip buffer
// at temporal offset `off` (telescoped skip sum). Bias + previous skip folded
// into the WMMA C initializer.
// ---------------------------------------------------------------------------
__global__ void __launch_bounds__(256) k_skip(
    const float* __restrict__ G, float* __restrict__ S,
    const float* __restrict__ Wk, const float* __restrict__ bk,
    int off, int initFlag) {
  int wid  = (blockIdx.x * blockDim.x + threadIdx.x) >> 5;
  int lane = threadIdx.x & 31;
  int half = lane >> 4;
  int l15  = lane & 15;
  int otile = wid & 1;
  int rtile = wid >> 1;
  int m0 = rtile << 4;
  int o  = (otile << 4) + l15;

  const float* Arow = G + (size_t)(m0 + l15 + off) * 32;
  float bias = bk[o];
  v8f c;
#pragma unroll
  for (int r = 0; r < 8; ++r) {
    int row = m0 + r + half * 8;
    float prev = initFlag ? 0.0f : S[(size_t)row * 32 + o];
    c[r] = bias + prev;
  }
#pragma unroll
  for (int k0 = 0; k0 < 32; k0 += 4) {
    int k = k0 + half * 2;
    v2f a = *(const v2f*)(Arow + k);
    v2f b = *(const v2f*)(Wk + (size_t)o * 32 + k);
    c = wmma_f32(a, b, c);
  }
#pragma unroll
  for (int r = 0; r < 8; ++r) {
    int row = m0 + r + half * 8;
    int l = row & 63;
    if (l < 61) S[(size_t)row * 32 + o] = c[r];
  }
}

// ---------------------------------------------------------------------------
// Kernel 4: graph diffusion Xout[b,m,j] = sum_n adj[n,m] * Xin[b,n,j].
// Per batch: GEMM adj^T(512x512) x X(512x2048).
// Block = 8 waves sharing one (b, m0): adj^T[m0:m0+16, 0:512] staged in LDS
// (pitch 516 floats -> bank = (4*m + k) & 63, conflict-free b64 reads).
// Each wave owns a 16x64 output tile: 4 accumulators share one LDS A tile.
// ---------------------------------------------------------------------------
#define LDSPITCH 516

__global__ void __launch_bounds__(256) k_diffuse(
    const float* __restrict__ adj, const float* __restrict__ Xin,
    float* __restrict__ Xout) {
  __shared__ float aT[16 * LDSPITCH];   // 33 KB of the 320 KB/WGP LDS

  int bid = blockIdx.x;         // 8(b) x 32(mt) x 4(cg) = 1024 blocks
  int cg = bid & 3;
  int mt = (bid >> 2) & 31;
  int b  = bid >> 7;
  int m0 = mt << 4;

  // Cooperative fill: aT[ml][k] = adj[k][m0+ml]  (16-lane 64B segments)
#pragma unroll 4
  for (int idx = threadIdx.x; idx < 8192; idx += 256) {
    int ml = idx & 15;
    int k  = idx >> 4;
    aT[ml * LDSPITCH + k] = adj[(size_t)k * 512 + m0 + ml];
  }
  __syncthreads();

  int wave = threadIdx.x >> 5;
  int lane = threadIdx.x & 31;
  int half = lane >> 4;
  int l15  = lane & 15;
  int j0   = (cg << 9) + (wave << 6);

  const float* Xb = Xin + (size_t)b * 512 * 2048;
  float* Ob       = Xout + (size_t)b * 512 * 2048;
  const float* Ar = aT + l15 * LDSPITCH;   // this lane's A row (M = l15)

  v8f c0 = {}, c1 = {}, c2 = {}, c3 = {};
#pragma unroll 4
  for (int n0 = 0; n0 < 512; n0 += 4) {
    int k = n0 + half * 2;
    v2f a = *(const v2f*)(Ar + k);          // single ds_load_b64
    const float* xr = Xb + (size_t)k * 2048 + j0 + l15;
    v2f b0, b1, b2, b3;
    b0.x = xr[0];  b0.y = xr[2048];
    b1.x = xr[16]; b1.y = xr[2048 + 16];
    b2.x = xr[32]; b2.y = xr[2048 + 32];
    b3.x = xr[48]; b3.y = xr[2048 + 48];
    c0 = wmma_f32(a, b0, c0);
    c1 = wmma_f32(a, b1, c1);
    c2 = wmma_f32(a, b2, c2);
    c3 = wmma_f32(a, b3, c3);
  }
#pragma unroll
  for (int r = 0; r < 8; ++r) {
    size_t orow = (size_t)(m0 + r + half * 8) * 2048 + j0 + l15;
    Ob[orow]      = c0[r];
    Ob[orow + 16] = c1[r];
    Ob[orow + 32] = c2[r];
    Ob[orow + 48] = c3[r];
  }
}

// ---------------------------------------------------------------------------
// Kernel 5: graph 1x1 conv over concat[h,x1,x2] (K=96) + BN + residual + BN.
// ---------------------------------------------------------------------------
__global__ void __launch_bounds__(256) k_gconv(
    const float* __restrict__ G, const float* __restrict__ X1,
    const float* __restrict__ X2, const float* __restrict__ Hres,
    float* __restrict__ Hout, const float* __restrict__ Wgc,
    const float* __restrict__ bgc, const float* __restrict__ gcg,
    const float* __restrict__ gcb, const float* __restrict__ bng,
    const float* __restrict__ bnb, int Lout) {
  int wid  = (blockIdx.x * blockDim.x + threadIdx.x) >> 5;
  int lane = threadIdx.x & 31;
  int half = lane >> 4;
  int l15  = lane & 15;
  int otile = wid & 1;
  int rtile = wid >> 1;
  int m0 = rtile << 4;
  int o  = (otile << 4) + l15;

  size_t arow = (size_t)(m0 + l15) * 32;
  float bias = bgc[o];
  v8f c;
#pragma unroll
  for (int r = 0; r < 8; ++r) c[r] = bias;

  const float* srcs[3] = {G, X1, X2};
#pragma unroll
  for (int part = 0; part < 3; ++part) {
    const float* src = srcs[part];
#pragma unroll
    for (int k0 = 0; k0 < 32; k0 += 4) {
      int k = k0 + half * 2;
      v2f a = *(const v2f*)(src + arow + k);
      v2f b = *(const v2f*)(Wgc + (size_t)o * 96 + part * 32 + k);
      c = wmma_f32(a, b, c);
    }
  }
  float rinv = rsqrtf(1.0f + EPSBN);
  float ig = gcg[o] * rinv, bgv = gcb[o];
  float ib = bng[o] * rinv, bbv = bnb[o];
#pragma unroll
  for (int r = 0; r < 8; ++r) {
    int row = m0 + r + half * 8;
    int l = row & 63;
    if (l < Lout) {
      float g = c[r] * ig + bgv;                         // gcbn
      float h = g + Hres[(size_t)(row + 1) * 32 + o];    // + residual[l+1]
      Hout[(size_t)row * 32 + o] = h * ib + bbv;         // bn
    }
  }
}

// ---------------------------------------------------------------------------
// Kernel 6: output BN + transpose to [B, 61, N, C].
// ---------------------------------------------------------------------------
__global__ void k_out(const float* __restrict__ S, const float* __restrict__ og,
                      const float* __restrict__ ob, float* __restrict__ out) {
  int t = blockIdx.x * blockDim.x + threadIdx.x;
  if (t >= 8 * 61 * 512 * 32) return;
  int cch = t & 31;
  int n   = (t >> 5) & 511;
  int rem = t >> 14;      // b*61 + l
  int l   = rem % 61;
  int b   = rem / 61;
  float inv = rsqrtf(1.0f + EPSBN);
  float v = S[(((size_t)(b * 512 + n)) * 64 + l) * 32 + cch];
  out[t] = v * (og[cch] * inv) + ob[cch];
}

// ---------------------------------------------------------------------------
extern "C" void kernel_launch(void* const* d_in, const int* in_sizes, int n_in,
                              void* d_out, int out_size, void* d_ws, size_t ws_size,
                              hipStream_t stream) {
  (void)in_sizes; (void)n_in; (void)out_size; (void)ws_size;
  const float* x   = (const float*)d_in[0];
  const float* adj = (const float*)d_in[1];
  const float* Wst = (const float*)d_in[2];
  const float* bst = (const float*)d_in[3];
  const float* Wf  = (const float*)d_in[4];
  const float* bf  = (const float*)d_in[5];
  const float* Wg  = (const float*)d_in[6];
  const float* bg  = (const float*)d_in[7];
  const float* Wsk = (const float*)d_in[8];
  const float* bsk = (const float*)d_in[9];
  const float* Wgc = (const float*)d_in[10];
  const float* bgc = (const float*)d_in[11];
  const float* gcg = (const float*)d_in[12];
  const float* gcb = (const float*)d_in[13];
  const float* bng = (const float*)d_in[14];
  const float* bnb = (const float*)d_in[15];
  const float* og  = (const float*)d_in[16];
  const float* ob  = (const float*)d_in[17];
  float* out = (float*)d_out;

  // Workspace: six [8][512][64][32] fp32 buffers (+slack for tap/offset reads).
  const size_t BUF = (size_t)8 * 512 * 64 * 32 + 1024;
  float* ws    = (float*)d_ws;
  float* bufH  = ws;
  float* bufH2 = ws + BUF;
  float* bufG  = ws + 2 * BUF;
  float* bufX1 = ws + 3 * BUF;
  float* bufX2 = ws + 4 * BUF;
  float* bufS  = ws + 5 * BUF;

  k_start<<<32768, 256, 0, stream>>>(x, Wst, bst, bufH);

  float* H  = bufH;
  float* Hn = bufH2;
  for (int i = 0; i < 3; ++i) {
    int Lout = 63 - i;
    k_gated<<<4096, 256, 0, stream>>>(H, bufG, Wf + i * 2048, bf + i * 32,
                                      Wg + i * 2048, bg + i * 32, Lout);
    k_skip<<<4096, 256, 0, stream>>>(bufG, bufS, Wsk + i * 1024, bsk + i * 32,
                                     2 - i, i == 0 ? 1 : 0);
    if (i < 2) {
      k_diffuse<<<1024, 256, 0, stream>>>(adj, bufG, bufX1);
      k_diffuse<<<1024, 256, 0, stream>>>(adj, bufX1, bufX2);
      k_gconv<<<4096, 256, 0, stream>>>(bufG, bufX1, bufX2, H, Hn,
                                        Wgc + i * 3072, bgc + i * 32,
                                        gcg + i * 32, gcb + i * 32,
                                        bng + i * 32, bnb + i * 32, Lout);
      float* tmp = H; H = Hn; Hn = tmp;
    }
  }
  k_out<<<31232, 256, 0, stream>>>(bufS, og, ob, out);
}